// MultiScaleDeformableAttention_23235773071472
// MI455X (gfx1250) — compile-verified
//
#include <hip/hip_runtime.h>
#include <hip/hip_bf16.h>
#include <stdint.h>

// Problem constants (static metadata in the reference)
#define B_ 4
#define S_ 21760
#define H_ 8
#define D_ 32
#define Q_ 10000
#define QPB 128   // queries per block
#define QPW 16    // queries per wave (8 waves/block)

// Levels 2 and 3 (32x32 and 16x16) staged in LDS: s in [20480, 21760)
#define STAGE_S0 20480
#define STAGE_NS 1280
#define LDS_FLOATS (STAGE_NS * D_)   // 40960 floats = 160 KB

#if defined(__has_builtin)
# if __has_builtin(__builtin_amdgcn_tensor_load_to_lds) && \
     __has_builtin(__builtin_amdgcn_s_wait_tensorcnt)
#  define HAS_TDM 1
# endif
# if __has_builtin(__builtin_amdgcn_global_load_async_to_lds_b128)
#  define HAS_ASYNC_LOAD 1
# endif
# if __has_builtin(__builtin_amdgcn_s_wait_asynccnt)
#  define HAS_ASYNC_WAIT 1
# endif
#endif

typedef int  v4i  __attribute__((ext_vector_type(4)));
typedef unsigned int u32x4 __attribute__((ext_vector_type(4)));
typedef int  i32x4 __attribute__((ext_vector_type(4)));
typedef int  i32x8 __attribute__((ext_vector_type(8)));
typedef __attribute__((address_space(1))) v4i* as1_v4i_p;
typedef __attribute__((address_space(3))) v4i* as3_v4i_p;

__device__ __forceinline__ as1_v4i_p to_as1_v4i(const void* p) {
  return (as1_v4i_p)(uintptr_t)p;  // 64-bit flat address == AS1 address
}
__device__ __forceinline__ as3_v4i_p to_as3_v4i(void* p) {
  return (as3_v4i_p)(uint32_t)(uintptr_t)p;  // low 32 bits = LDS offset
}

__global__ void msda_fwd(const float* __restrict__ value,
                         const float* __restrict__ loc,
                         const float* __restrict__ attw,
                         float* __restrict__ out) {
  constexpr int lvlW_[4] = {128, 64, 32, 16};
  constexpr int lvlH_[4] = {128, 64, 32, 16};
  constexpr int lvlS_[4] = {0, 16384, 20480, 21504};

  extern __shared__ float lds[];  // 160 KB dynamic

  const int b    = blockIdx.z;
  const int h    = blockIdx.y;
  const int tid  = threadIdx.x;
  const int lane = tid & 31;
  const int wave = tid >> 5;

  // ---- Stage levels 2+3 of value[b, :, h, :] into LDS ----
  {
    const float* gbase =
        value + ((size_t)((size_t)b * S_ + STAGE_S0) * H_ + h) * D_;
#if defined(HAS_TDM)
    // One TDM descriptor: 1280-row x 128B tile, global row stride 1KB -> LDS.
    if (wave == 0) {
      const uint64_t gaddr = (uint64_t)(uintptr_t)gbase;
      const uint32_t laddr = (uint32_t)(uintptr_t)&lds[0];
      u32x4 g0;
      g0.x = 0x1u;                                   // count=1, user mode
      g0.y = laddr;                                  // lds_addr [63:32]
      g0.z = (uint32_t)(gaddr & 0xffffffffu);        // global_addr lo
      g0.w = (uint32_t)((gaddr >> 32) & 0x01ffffffu) // global_addr [56:32]
             | (2u << 30);                           // type=2 ("image")
      i32x8 g1;
      g1[0] = (int)(2u << 16);          // data_size=4B; wg_mask=0; no pad
      g1[1] = (int)((uint32_t)D_ << 16);        // tensor_dim0[15:0]=32
      g1[2] = (int)((uint32_t)STAGE_NS << 16);  // dim0 hi=0 | tensor_dim1 lo=1280
      g1[3] = (int)((uint32_t)D_ << 16);        // tensor_dim1 hi=0 | tile_dim0=32
      g1[4] = (int)(uint32_t)STAGE_NS;          // tile_dim1=1280 | tile_dim2=0
      g1[5] = (int)(uint32_t)(H_ * D_);         // tensor_dim0_stride=256 elems
      g1[6] = 0;                                // stride hi | dim1_stride lo
      g1[7] = 0;                                // dim1_stride hi
      i32x4 gz4 = {0, 0, 0, 0};                 // groups 2-3: unused (2-D tile)
      i32x8 gz8 = {0, 0, 0, 0, 0, 0, 0, 0};     // trailing group: unused
      __builtin_amdgcn_tensor_load_to_lds(g0, g1, gz4, gz4, gz8, 0);
      __builtin_amdgcn_s_wait_tensorcnt(0);
    }
#else
    // 1280 rows x 128B; chunk c -> row r = c>>3, 16B group sub = c&7
    for (int c = tid; c < STAGE_NS * 8; c += blockDim.x) {
      const int r = c >> 3, sub = c & 7;
      const float* gp = gbase + (size_t)r * (H_ * D_) + sub * 4;
      float* lp = &lds[r * D_ + sub * 4];
# if defined(HAS_ASYNC_LOAD)
      __builtin_amdgcn_global_load_async_to_lds_b128(to_as1_v4i(gp),
                                                     to_as3_v4i(lp), 0, 0);
# else
      *(float4*)lp = *(const float4*)gp;
# endif
    }
# if defined(HAS_ASYNC_LOAD) && defined(HAS_ASYNC_WAIT)
    __builtin_amdgcn_s_wait_asynccnt(0);
# endif
#endif
    __syncthreads();
  }

  const int qBase = blockIdx.x * QPB + wave * QPW;
  // value[b, s, h, lane] = vbase[s * 256]
  const float* vbase = value + ((size_t)b * S_ * H_ + h) * D_ + lane;

  for (int j = 0; j < QPW; ++j) {
    const int q = qBase + j;
    if (q >= Q_) return;  // wave-uniform; no barriers after this point
    const size_t qh = ((size_t)b * Q_ + q) * H_ + h;
    const float* locp = loc + qh * 32;   // [L,P,2] = 32 floats
    const float* awp  = attw + qh * 16;  // [L,P]   = 16 floats

    if (j + 1 < QPW && q + 1 < Q_) {
      __builtin_prefetch(locp + H_ * 32, 0, 0);  // next query's locations
      __builtin_prefetch(awp + H_ * 16, 0, 0);   // next query's weights
    }

    // Streamed once -> non-temporal, keep `value` resident in L2
    const float locv = __builtin_nontemporal_load(locp + lane);
    const float awv  = (lane < 16) ? __builtin_nontemporal_load(awp + lane) : 0.f;
    float acc = 0.f;

#pragma unroll
    for (int i = 0; i < 16; ++i) {
      const int l  = i >> 2;
      const int W  = lvlW_[l], Hh = lvlH_[l], SS = lvlS_[l];
      const float x01 = __shfl(locv, 2 * i, 32);
      const float y01 = __shfl(locv, 2 * i + 1, 32);
      const float aw  = __shfl(awv, i, 32);

      // pixel coords: (2*loc-1 + 1)*dim/2 - 0.5 = loc*dim - 0.5
      const float x = x01 * (float)W - 0.5f;
      const float y = y01 * (float)Hh - 0.5f;
      const float xf = floorf(x), yf = floorf(y);
      const float fx = x - xf, fy = y - yf;
      const int x0 = (int)xf, y0 = (int)yf;
      const int x1 = x0 + 1,  y1 = y0 + 1;

      const bool vx0 = (x0 >= 0) & (x0 < W);
      const bool vx1 = (x1 >= 0) & (x1 < W);
      const bool vy0 = (y0 >= 0) & (y0 < Hh);
      const bool vy1 = (y1 >= 0) & (y1 < Hh);
      const int xc0 = min(max(x0, 0), W - 1);
      const int xc1 = min(max(x1, 0), W - 1);
      const int yc0 = min(max(y0, 0), Hh - 1);
      const int yc1 = min(max(y1, 0), Hh - 1);

      const float w00 = (vx0 && vy0) ? (1.f - fx) * (1.f - fy) * aw : 0.f;
      const float w01 = (vx1 && vy0) ? fx * (1.f - fy) * aw : 0.f;
      const float w10 = (vx0 && vy1) ? (1.f - fx) * fy * aw : 0.f;
      const float w11 = (vx1 && vy1) ? fx * fy * aw : 0.f;

      const int s00 = SS + yc0 * W + xc0;
      const int s01 = SS + yc0 * W + xc1;
      const int s10 = SS + yc1 * W + xc0;
      const int s11 = SS + yc1 * W + xc1;

      float v00, v01, v10, v11;
      if (l < 2) {
        // L2-resident levels: one aligned 128B line per corner
        v00 = vbase[(size_t)s00 * (H_ * D_)];
        v01 = vbase[(size_t)s01 * (H_ * D_)];
        v10 = vbase[(size_t)s10 * (H_ * D_)];
        v11 = vbase[(size_t)s11 * (H_ * D_)];
      } else {
        // LDS-staged levels: bank-conflict-free (lane = consecutive dword)
        v00 = lds[(s00 - STAGE_S0) * D_ + lane];
        v01 = lds[(s01 - STAGE_S0) * D_ + lane];
        v10 = lds[(s10 - STAGE_S0) * D_ + lane];
        v11 = lds[(s11 - STAGE_S0) * D_ + lane];
      }
      acc = fmaf(w00, v00, acc);
      acc = fmaf(w01, v01, acc);
      acc = fmaf(w10, v10, acc);
      acc = fmaf(w11, v11, acc);
    }
    // out[b, q, h*D + d] — streamed, non-temporal
    __builtin_nontemporal_store(acc, out + qh * D_ + lane);
  }
}

extern "C" void kernel_launch(void* const* d_in, const int* in_sizes, int n_in,
                              void* d_out, int out_size, void* d_ws, size_t ws_size,
                              hipStream_t stream) {
  (void)in_sizes; (void)n_in; (void)out_size; (void)d_ws; (void)ws_size;
  const float* value = (const float*)d_in[0];
  // d_in[1] spatial shapes, d_in[2] level starts: static metadata, hardcoded
  const float* loc  = (const float*)d_in[3];
  const float* attw = (const float*)d_in[4];
  // d_in[5] im2col_step: unused
  float* out = (float*)d_out;

  const size_t shmem = (size_t)LDS_FLOATS * sizeof(float);  // 160 KB
  hipError_t err = hipFuncSetAttribute(
      (const void*)msda_fwd, hipFuncAttributeMaxDynamicSharedMemorySize,
      (int)shmem);
  (void)err;
  dim3 grid((Q_ + QPB - 1) / QPB, H_, B_);
  msda_fwd<<<grid, dim3(256), shmem, stream>>>(value, loc, attw, out);
}